// BertCRF_80427557585285
// MI455X (gfx1250) — compile-verified
//
#include <hip/hip_runtime.h>
#include <hip/hip_bf16.h>
#include <math.h>

typedef float v2f __attribute__((ext_vector_type(2)));
typedef float v8f __attribute__((ext_vector_type(8)));

#define B_SZ   64
#define T_SZ   512
#define H_SZ   768
#define L_SZ   25
#define LP     32                 // labels padded to 32 (two 16-wide WMMA N tiles)
#define M_ROWS (B_SZ * T_SZ)      // 32768 rows
#define M_TILES (M_ROWS / 16)     // 2048 16-row tiles
#define KCHUNK 256                // K staged through LDS in chunks
#define SB_STRIDE 36              // LDS row stride (multiple of 4 for aligned b128 fills)

// ---- workspace layout (in floats) ----
#define WS_WPAD 0                         // 768*32 padded weights
#define WS_BPAD (H_SZ * LP)               // 32 padded bias
#define WS_EMIS (WS_BPAD + LP)            // 32768*32 padded emissions
#define WS_LLH  (WS_EMIS + M_ROWS * LP)   // 64 per-batch log-likelihoods

// ------------------------------------------------------------------
// Pad W_em [768,25] -> [768,32] (zeros) and b_em [25] -> [32]
// ------------------------------------------------------------------
__global__ __launch_bounds__(256) void pad_kernel(const float* __restrict__ W,
                                                  const float* __restrict__ bvec,
                                                  float* __restrict__ Wp,
                                                  float* __restrict__ bp) {
    int idx = blockIdx.x * blockDim.x + threadIdx.x;
    if (idx < H_SZ * LP) {
        int h = idx >> 5;
        int l = idx & 31;
        Wp[idx] = (l < L_SZ) ? W[h * L_SZ + l] : 0.0f;
    }
    if (idx < LP) bp[idx] = (idx < L_SZ) ? bvec[idx] : 0.0f;
}

// ------------------------------------------------------------------
// Emissions GEMM: emis[32768,32] = X[32768,768] * Wp[768,32] + bp
// One wave per 16x32 output tile, V_WMMA_F32_16X16X4_F32, K staged
// through LDS in 256-deep chunks.
// ------------------------------------------------------------------
__global__ __launch_bounds__(256) void gemm_emis(const float* __restrict__ X,
                                                 const float* __restrict__ Wp,
                                                 const float* __restrict__ bp,
                                                 float* __restrict__ emis) {
    __shared__ float sB[KCHUNK * SB_STRIDE];   // 36 KB

    const int tid   = threadIdx.x;
    const int lane  = tid & 31;
    const int gwave = blockIdx.x * 8 + (tid >> 5);   // 0..1023 (128 blocks * 8 waves)
    const int n     = lane & 15;                     // N within tile / M row within tile
    const int hi    = lane >> 4;                     // K-half selector for A/B fragments

    const float bias0 = bp[n];
    const float bias1 = bp[16 + n];

    for (int rep = 0; rep < 2; ++rep) {              // 1024 waves * 2 = 2048 tiles
        const int tile = rep * 1024 + gwave;
        v8f c0 = {0.f, 0.f, 0.f, 0.f, 0.f, 0.f, 0.f, 0.f};
        v8f c1 = {0.f, 0.f, 0.f, 0.f, 0.f, 0.f, 0.f, 0.f};

        // lane's A row: M = tile*16 + (lane&15); elements k+2*hi, k+2*hi+1
        const float* aRow = X + (tile * 16 + n) * H_SZ;

        for (int chunk = 0; chunk < 3; ++chunk) {
            __syncthreads();   // protect sB from previous iteration's readers
            // cooperative fill: 256 rows x 32 cols of Wp -> sB (16B-aligned stores)
            for (int e = tid * 4; e < KCHUNK * LP; e += 256 * 4) {
                const float4 w = *(const float4*)(Wp + chunk * (KCHUNK * LP) + e);
                float* dst = sB + (e >> 5) * SB_STRIDE + (e & 31);
                dst[0] = w.x; dst[1] = w.y; dst[2] = w.z; dst[3] = w.w;
            }
            __syncthreads();

            const float* aPtr = aRow + chunk * KCHUNK + 2 * hi;
            #pragma unroll 4
            for (int k = 0; k < KCHUNK; k += 4) {
                v2f a;
                a.x = aPtr[k];
                a.y = aPtr[k + 1];

                const int kb = (k + 2 * hi) * SB_STRIDE;
                v2f b0, b1;
                b0.x = sB[kb + n];
                b0.y = sB[kb + SB_STRIDE + n];
                b1.x = sB[kb + 16 + n];
                b1.y = sB[kb + SB_STRIDE + 16 + n];

                c0 = __builtin_amdgcn_wmma_f32_16x16x4_f32(
                        false, a, false, b0, (short)0, c0, false, false);
                c1 = __builtin_amdgcn_wmma_f32_16x16x4_f32(
                        false, a, false, b1, (short)0, c1, false, false);
            }
        }

        // C/D layout: VGPR r -> (M = r + 8*hi, N = lane&15 [+16 for c1])
        float* out = emis + (tile * 16 + 8 * hi) * LP + n;
        #pragma unroll
        for (int r = 0; r < 8; ++r) {
            out[r * LP]      = c0[r] + bias0;
            out[r * LP + 16] = c1[r] + bias1;
        }
    }
}

// ------------------------------------------------------------------
// CRF: forward algorithm in scaled-probability space + numerator.
// One wave32 per batch element; lane j = state j (j >= 25 padded out).
// ------------------------------------------------------------------
__global__ __launch_bounds__(32) void crf_kernel(const float* __restrict__ emis,
                                                 const int*   __restrict__ mask,
                                                 const int*   __restrict__ labels,
                                                 const float* __restrict__ trans,
                                                 const float* __restrict__ start_t,
                                                 const float* __restrict__ end_t,
                                                 float* __restrict__ llh) {
    __shared__ float sA[32];
    const int  b   = blockIdx.x;
    const int  j   = threadIdx.x;
    const bool act = (j < L_SZ);
    const int  base = b * T_SZ;

    // exp of transition column j, kept in registers
    float expT[L_SZ];
    #pragma unroll
    for (int i = 0; i < L_SZ; ++i) {
        const float tv = act ? trans[i * L_SZ + j] : 0.0f;
        expT[i] = expf(tv);
    }

    // alpha0 = start + emissions[t=0]; convert to scaled-prob space
    float a0 = act ? (start_t[j] + emis[base * LP + j]) : -1e30f;
    float mx = a0;
    #pragma unroll
    for (int d = 16; d; d >>= 1) mx = fmaxf(mx, __shfl_xor(mx, d, 32));
    float alphaP = act ? expf(a0 - mx) : 0.0f;
    float logZ = mx;

    for (int t = 1; t < T_SZ; ++t) {
        sA[j] = alphaP;
        __syncthreads();
        float acc = 0.0f;
        #pragma unroll
        for (int i = 0; i < L_SZ; ++i) acc = fmaf(sA[i], expT[i], acc);
        __syncthreads();

        if (mask[base + t] != 0) {             // uniform branch per wave
            const float em  = act ? emis[(base + t) * LP + j] : 0.0f;
            const float nxt = acc * (act ? expf(em) : 0.0f);
            float m2 = nxt;
            #pragma unroll
            for (int d = 16; d; d >>= 1) m2 = fmaxf(m2, __shfl_xor(m2, d, 32));
            alphaP = nxt / m2;
            logZ += logf(m2);
        }
    }

    // denominator: logZ + log(sum_j alphaP_j * exp(end_t[j]))
    float s = act ? alphaP * expf(end_t[j]) : 0.0f;
    #pragma unroll
    for (int d = 16; d; d >>= 1) s += __shfl_xor(s, d, 32);
    const float denom = logZ + logf(s);

    // numerator: lane-strided over t, wave reduce
    float num = 0.0f;
    int msum = 0;
    for (int t = j; t < T_SZ; t += 32) {
        const int m   = mask[base + t];
        const int tag = labels[base + t];
        msum += m;
        if (t == 0) {
            num += start_t[tag] + emis[base * LP + tag];
        } else if (m != 0) {
            const int prev = labels[base + t - 1];
            num += trans[prev * L_SZ + tag] + emis[(base + t) * LP + tag];
        }
    }
    #pragma unroll
    for (int d = 16; d; d >>= 1) {
        num  += __shfl_xor(num, d, 32);
        msum += __shfl_xor(msum, d, 32);
    }

    if (j == 0) {
        const int lastIdx = msum - 1;
        const int lt = labels[base + lastIdx];
        llh[b] = (num + end_t[lt]) - denom;
    }
}

// ------------------------------------------------------------------
// Final: out = -mean(llh)
// ------------------------------------------------------------------
__global__ __launch_bounds__(32) void reduce_kernel(const float* __restrict__ llh,
                                                    float* __restrict__ out) {
    const int j = threadIdx.x;
    float s = llh[j] + llh[j + 32];
    #pragma unroll
    for (int d = 16; d; d >>= 1) s += __shfl_xor(s, d, 32);
    if (j == 0) out[0] = -s * (1.0f / (float)B_SZ);
}

extern "C" void kernel_launch(void* const* d_in, const int* in_sizes, int n_in,
                              void* d_out, int out_size, void* d_ws, size_t ws_size,
                              hipStream_t stream) {
    (void)in_sizes; (void)n_in; (void)out_size; (void)ws_size;

    const float* X       = (const float*)d_in[0];  // encoder_logits [64,512,768]
    const int*   mask    = (const int*)  d_in[1];  // attention_mask [64,512]
    const int*   labels  = (const int*)  d_in[2];  // labels         [64,512]
    const float* W       = (const float*)d_in[3];  // W_em           [768,25]
    const float* bvec    = (const float*)d_in[4];  // b_em           [25]
    const float* trans   = (const float*)d_in[5];  // trans          [25,25]
    const float* start_t = (const float*)d_in[6];  // start_t        [25]
    const float* end_t   = (const float*)d_in[7];  // end_t          [25]

    float* ws   = (float*)d_ws;
    float* Wp   = ws + WS_WPAD;
    float* bp   = ws + WS_BPAD;
    float* emis = ws + WS_EMIS;
    float* llh  = ws + WS_LLH;

    pad_kernel<<<(H_SZ * LP + 255) / 256, 256, 0, stream>>>(W, bvec, Wp, bp);
    gemm_emis<<<128, 256, 0, stream>>>(X, Wp, bp, emis);
    crf_kernel<<<B_SZ, 32, 0, stream>>>(emis, mask, labels, trans, start_t, end_t, llh);
    reduce_kernel<<<1, 32, 0, stream>>>(llh, (float*)d_out);
}